// RFDetModule_70669391888764
// MI455X (gfx1250) — compile-verified
//
#include <hip/hip_runtime.h>
#include <stdint.h>

typedef uint32_t u32;
typedef uint64_t u64;

// Problem constants (match reference setup_inputs / constants)
#define B_IMG 8
#define H_IMG 2048
#define W_IMG 2048
#define NPIX  (H_IMG * W_IMG)
#define BORDER 8
#define NMS_R 2          // 5x5 NMS
#define TOPK 512
#define GR 7             // 15x15 gaussian, sigma = 0.5 -> w(d)=exp(-2*d^2)

// Tiling for the NMS pass
#define TW 64
#define TH 32
#define LW (TW + 2 * NMS_R)   // 68
#define LH (TH + 2 * NMS_R)   // 36
#define RPT (TH / 4)          // 8 rows per thread (256 threads, 64 wide)

// Workspace capacities
#define MAXC   (1u << 18)     // candidate slots per image (~165k expected survivors)
#define SELCAP 1024
#define EQCAP  64

// ---- CDNA5 TDM availability (6-arg form per this toolchain's headers) ----
#if defined(__gfx1250__) && defined(__HIP_DEVICE_COMPILE__)
#  if __has_builtin(__builtin_amdgcn_tensor_load_to_lds) && \
      __has_builtin(__builtin_amdgcn_s_wait_tensorcnt)
#    define HAVE_TDM 1
#  endif
#endif
#ifndef HAVE_TDM
#  define HAVE_TDM 0
#endif

typedef unsigned int v4u __attribute__((ext_vector_type(4)));
typedef int          v8i __attribute__((ext_vector_type(8)));
typedef int          v4i __attribute__((ext_vector_type(4)));
typedef float        v4f __attribute__((ext_vector_type(4)));

// Wave32 ballot-aggregated candidate append (1 atomic per wave instead of 32).
__device__ __forceinline__ void push_cand(u64* __restrict__ candImg,
                                          u32* __restrict__ cnt,
                                          bool pred, u64 key) {
  unsigned long long msk = __ballot(pred);
  unsigned lane = threadIdx.x & 31u;
  int nsel = __popcll(msk);
  int base = 0;
  if (lane == 0 && nsel) base = (int)atomicAdd(cnt, (u32)nsel);
  base = __shfl(base, 0, 32);
  if (pred) {
    u32 pos = (u32)base + (u32)__popcll(msk & ((1ull << lane) - 1ull));
    if (pos < MAXC) candImg[pos] = key;
  }
}

// -------- Kernel 0: zero the per-call counters in workspace --------
__global__ void init_counters(u32* __restrict__ ctr) {
  if (threadIdx.x < 3 * B_IMG) ctr[threadIdx.x] = 0;  // cand_count, sel_count, eq_cnt
}

// -------- Kernel 0b: zero out_x and out_mask regions (streaming, NT) --------
__global__ void zero_out(v4f* __restrict__ p, size_t n4) {
  size_t i = (size_t)blockIdx.x * blockDim.x + threadIdx.x;
  size_t stride = (size_t)gridDim.x * blockDim.x;
  v4f z = {0.f, 0.f, 0.f, 0.f};
  for (; i < n4; i += stride) __builtin_nontemporal_store(z, &p[i]);
}

// -------- Kernel 1: border mask + 5x5 NMS -> topk_value + candidate list --------
__global__ __launch_bounds__(256) void nms_tv_kernel(const float* __restrict__ in,
                                                     float* __restrict__ out_tv,
                                                     u64* __restrict__ cand,
                                                     u32* __restrict__ cand_count) {
  __shared__ float tile[LW * LH];
  const int c0 = blockIdx.x * TW;
  const int r0 = blockIdx.y * TH;
  const int img = blockIdx.z;
  const int gx0 = c0 - NMS_R;
  const int gy0 = r0 - NMS_R;
  const float* src = in + (size_t)img * NPIX;
  // Boundary tiles are exactly the ones whose halo leaves the image AND the
  // only ones intersecting the 8-px zero border (since TH,TW > BORDER+NMS_R).
  const bool edge = (r0 == 0) || (c0 == 0) || (r0 + TH == H_IMG) || (c0 + TW == W_IMG);

#if HAVE_TDM
  if (!edge) {
    // Interior tile: one TDM 2-D tile DMA (global -> LDS), issued by wave 0.
    if (threadIdx.x < 32u) {
      u64 gaddr = (u64)(uintptr_t)(src + (size_t)gy0 * W_IMG + gx0);
      u32 ldsoff = (u32)(uintptr_t)(&tile[0]);          // low 32b = LDS byte offset
      u32 td0 = (u32)(W_IMG - gx0);                     // remaining extent from tile start
      u32 td1 = (u32)(H_IMG - gy0);
      v4u g0;
      g0.x = 1u;                                        // count=1, user descriptor
      g0.y = ldsoff;                                    // lds_addr
      g0.z = (u32)gaddr;                                // global_addr[31:0]
      g0.w = ((u32)(gaddr >> 32) & 0x01ffffffu) | 0x80000000u;  // addr[56:32] | type=2
      v8i g1;
      g1[0] = (int)(2u << 16);                          // wg_mask=0, data_size=2 (4B)
      g1[1] = (int)((td0 & 0xffffu) << 16);             // tensor_dim0[15:0]
      g1[2] = (int)((td0 >> 16) | ((td1 & 0xffffu) << 16));   // dim0 hi | dim1 lo
      g1[3] = (int)((td1 >> 16) | ((u32)LW << 16));     // dim1 hi | tile_dim0 = 68
      g1[4] = (int)LH;                                  // tile_dim1 = 36, tile_dim2 = 0
      g1[5] = (int)W_IMG;                               // tensor_dim0_stride lo = 2048
      g1[6] = 0;                                        // stride hi, dim1_stride lo
      g1[7] = 0;
      v4i gz4 = {0, 0, 0, 0};
      v8i gz8 = {0, 0, 0, 0, 0, 0, 0, 0};
      __builtin_amdgcn_tensor_load_to_lds(g0, g1, gz4, gz4, gz8, 0);
      __builtin_amdgcn_s_wait_tensorcnt(0);
    }
  } else
#endif
  {
    // Software staging (all tiles when TDM unavailable; boundary tiles otherwise).
    for (int i = threadIdx.x; i < LW * LH; i += 256) {
      int ly = i / LW, lx = i - ly * LW;
      int gy = gy0 + ly, gx = gx0 + lx;
      float v = 0.0f;
      if ((unsigned)gy < (unsigned)H_IMG && (unsigned)gx < (unsigned)W_IMG)
        v = src[(size_t)gy * W_IMG + gx];
      tile[i] = v;
    }
  }
  __syncthreads();

  if (edge) {  // apply the 8-px zero border inside the staged tile
    for (int i = threadIdx.x; i < LW * LH; i += 256) {
      int ly = i / LW, lx = i - ly * LW;
      int gy = gy0 + ly, gx = gx0 + lx;
      if (gy < BORDER || gy >= H_IMG - BORDER || gx < BORDER || gx >= W_IMG - BORDER)
        tile[i] = 0.0f;
    }
    __syncthreads();
  }

  // Separable sliding 5x5 max: 12 row-maxes feed 8 output rows per thread.
  const int tx = (int)(threadIdx.x & (TW - 1));
  const int trow = (int)(threadIdx.x >> 6);
  float rm[RPT + 4];
#pragma unroll
  for (int j = 0; j < RPT + 4; ++j) {
    const float* p = &tile[(trow * RPT + j) * LW + tx];
    rm[j] = fmaxf(fmaxf(fmaxf(p[0], p[1]), fmaxf(p[2], p[3])), p[4]);
  }
  float* tvp = out_tv + (size_t)img * NPIX;
  u64* candImg = cand + (size_t)img * MAXC;
  u32* cnt = cand_count + img;
#pragma unroll
  for (int k = 0; k < RPT; ++k) {
    float m = fmaxf(fmaxf(fmaxf(rm[k], rm[k + 1]), fmaxf(rm[k + 2], rm[k + 3])), rm[k + 4]);
    float c = tile[(trow * RPT + k + NMS_R) * LW + tx + NMS_R];
    float tv = (c >= m) ? c : 0.0f;   // nms mask * x  (ties -> keep, like reference)
    u32 pix = (u32)((r0 + trow * RPT + k) * W_IMG + (c0 + tx));
    __builtin_nontemporal_store(tv, tvp + pix);
    push_cand(candImg, cnt, tv > 0.0f, ((u64)__float_as_uint(tv) << 32) | (u64)pix);
  }
}

// -------- Kernel 2: 4-pass MSB radix select of the 512th-largest value --------
__global__ __launch_bounds__(256) void radix_select(const u64* __restrict__ cand,
                                                    const u32* __restrict__ cand_count,
                                                    u32* __restrict__ thresh,
                                                    u32* __restrict__ need_eq) {
  const int img = blockIdx.x;
  const u64* c = cand + (size_t)img * MAXC;
  u32 n = cand_count[img]; if (n > MAXC) n = MAXC;
  __shared__ u32 hist[256];
  __shared__ u32 sh_prefix, sh_k;
  if (n == 0) {
    if (threadIdx.x == 0) { thresh[img] = 0xFFFFFFFFu; need_eq[img] = 0; }
    return;
  }
  if (threadIdx.x == 0) { sh_prefix = 0; sh_k = (n < TOPK) ? n : TOPK; }
  for (int p = 0; p < 4; ++p) {
    hist[threadIdx.x] = 0;
    __syncthreads();
    const u32 prefix = sh_prefix;
    const u32 K = sh_k;
    const int sh_hi = 32 - 8 * p;
    const int sh_by = 24 - 8 * p;
    for (u32 i = threadIdx.x; i < n; i += 256) {
      u32 v = (u32)(c[i] >> 32);
      if (p == 0 || (v >> sh_hi) == prefix)
        atomicAdd(&hist[(v >> sh_by) & 255u], 1u);
    }
    __syncthreads();
    if (threadIdx.x == 0) {
      u32 cum = 0; int b = 0;
      for (int j = 255; j >= 0; --j) {
        u32 h = hist[j];
        if (cum + h >= K) { b = j; break; }
        cum += h;
      }
      sh_prefix = (prefix << 8) | (u32)b;
      sh_k = K - cum;                    // # of ==T still needed after ">T" ones
    }
    __syncthreads();
  }
  if (threadIdx.x == 0) { thresh[img] = sh_prefix; need_eq[img] = sh_k; }
}

// -------- Kernel 3: select >T, record ==T ties; scatter mask=1.0 --------
__global__ void select_mark(const u64* __restrict__ cand, const u32* __restrict__ cand_count,
                            const u32* __restrict__ thresh, u32* __restrict__ eq_cnt,
                            u32* __restrict__ eqlist, u64* __restrict__ sel,
                            u32* __restrict__ sel_count, float* __restrict__ out_mask) {
  const int img = blockIdx.y;
  const u64* c = cand + (size_t)img * MAXC;
  u32 n = cand_count[img]; if (n > MAXC) n = MAXC;
  const u32 T = thresh[img];
  const u32 stride = gridDim.x * blockDim.x;
  for (u32 i = blockIdx.x * blockDim.x + threadIdx.x; i < n; i += stride) {
    u64 key = c[i];
    u32 v = (u32)(key >> 32);
    if (v > T) {
      u32 s = atomicAdd(&sel_count[img], 1u);
      if (s < SELCAP) sel[(size_t)img * SELCAP + s] = key;
      out_mask[(size_t)img * NPIX + (u32)key] = 1.0f;
    } else if (v == T) {
      u32 e = atomicAdd(&eq_cnt[img], 1u);
      if (e < EQCAP) eqlist[img * EQCAP + e] = (u32)key;
    }
  }
}

// -------- Kernel 4: deterministic smallest-index tie-break for ==T --------
__global__ void finalize_eq(const u32* __restrict__ eqlist, const u32* __restrict__ eq_cnt,
                            const u32* __restrict__ thresh, const u32* __restrict__ need_eq,
                            u64* __restrict__ sel, u32* __restrict__ sel_count,
                            float* __restrict__ out_mask) {
  const int img = blockIdx.x;
  u32 m = eq_cnt[img]; if (m > EQCAP) m = EQCAP;
  u32 ne = need_eq[img]; if (ne > m) ne = m;
  const u32 t = threadIdx.x;
  if (t >= m) return;
  u32 mypix = eqlist[img * EQCAP + t];
  u32 rank = 0;
  for (u32 j = 0; j < m; ++j)
    if (eqlist[img * EQCAP + j] < mypix) ++rank;
  if (rank < ne) {                               // top_k breaks ties by lowest index
    u32 s = atomicAdd(&sel_count[img], 1u);
    if (s < SELCAP) sel[(size_t)img * SELCAP + s] = ((u64)thresh[img] << 32) | mypix;
    out_mask[(size_t)img * NPIX + mypix] = 1.0f;
  }
}

// -------- Kernel 5: sparse 15x15 gaussian scatter-add --------
__global__ __launch_bounds__(256) void gauss_scatter(const u64* __restrict__ sel,
                                                     const u32* __restrict__ sel_count,
                                                     float* __restrict__ out_x) {
  const int img = blockIdx.y;
  u32 ns = sel_count[img]; if (ns > SELCAP) ns = SELCAP;
  float* outp = out_x + (size_t)img * NPIX;
  const int t = (int)threadIdx.x;
  const int dy = t / 15 - GR, dx = t % 15 - GR;
  const float w = expf(-2.0f * (float)(dy * dy)) * expf(-2.0f * (float)(dx * dx));
  for (u32 s = blockIdx.x; s < ns; s += gridDim.x) {
    u64 key = sel[(size_t)img * SELCAP + s];
    u32 pix = (u32)key;
    float val = __uint_as_float((u32)(key >> 32));
    int py = (int)(pix / W_IMG), px = (int)(pix % W_IMG);
    if (t < 225) {
      int y = py + dy, x = px + dx;   // always in-bounds (points >= 8 from edge)
      if ((unsigned)y < (unsigned)H_IMG && (unsigned)x < (unsigned)W_IMG)
        atomicAdd(&outp[(size_t)y * W_IMG + x], w * val);
    }
  }
}

// -------- Kernel 6: clip [0,1] over the touched neighborhoods (idempotent) --------
__global__ __launch_bounds__(256) void gauss_clip(const u64* __restrict__ sel,
                                                  const u32* __restrict__ sel_count,
                                                  float* __restrict__ out_x) {
  const int img = blockIdx.y;
  u32 ns = sel_count[img]; if (ns > SELCAP) ns = SELCAP;
  float* outp = out_x + (size_t)img * NPIX;
  const int t = (int)threadIdx.x;
  const int dy = t / 15 - GR, dx = t % 15 - GR;
  for (u32 s = blockIdx.x; s < ns; s += gridDim.x) {
    u64 key = sel[(size_t)img * SELCAP + s];
    u32 pix = (u32)key;
    int py = (int)(pix / W_IMG), px = (int)(pix % W_IMG);
    if (t < 225) {
      int y = py + dy, x = px + dx;
      if ((unsigned)y < (unsigned)H_IMG && (unsigned)x < (unsigned)W_IMG) {
        size_t a = (size_t)y * W_IMG + x;
        float v = outp[a];
        outp[a] = fminf(fmaxf(v, 0.0f), 1.0f);
      }
    }
  }
}

extern "C" void kernel_launch(void* const* d_in, const int* in_sizes, int n_in,
                              void* d_out, int out_size, void* d_ws, size_t ws_size,
                              hipStream_t stream) {
  (void)in_sizes; (void)n_in; (void)out_size; (void)ws_size;
  const float* in = (const float*)d_in[0];
  float* out = (float*)d_out;
  float* out_x    = out;                                // conv result (output 1)
  float* out_mask = out + (size_t)B_IMG * NPIX;         // topk_mask as float (output 2)
  float* out_tv   = out + 2 * (size_t)B_IMG * NPIX;     // topk_value (output 3)

  // Workspace layout (~16.9 MB)
  char* ws = (char*)d_ws;
  u64* cand = (u64*)ws;                                               // B*MAXC u64
  size_t off = (size_t)B_IMG * MAXC * sizeof(u64);
  u64* sel = (u64*)(ws + off);                                        // B*SELCAP u64
  off += (size_t)B_IMG * SELCAP * sizeof(u64);
  u32* eqlist = (u32*)(ws + off);                                     // B*EQCAP u32
  off += (size_t)B_IMG * EQCAP * sizeof(u32);
  u32* ctr = (u32*)(ws + off);
  u32* cand_count = ctr;                // B   (zeroed each call)
  u32* sel_count  = ctr + B_IMG;        // B   (zeroed each call)
  u32* eq_cnt     = ctr + 2 * B_IMG;    // B   (zeroed each call)
  u32* thresh     = ctr + 3 * B_IMG;    // B   (written by radix_select)
  u32* need_eq    = ctr + 4 * B_IMG;    // B   (written by radix_select)

  init_counters<<<1, 64, 0, stream>>>(ctr);
  zero_out<<<8192, 256, 0, stream>>>((v4f*)out, (size_t)2 * B_IMG * NPIX / 4);

  dim3 g1(W_IMG / TW, H_IMG / TH, B_IMG);
  nms_tv_kernel<<<g1, 256, 0, stream>>>(in, out_tv, cand, cand_count);

  radix_select<<<B_IMG, 256, 0, stream>>>(cand, cand_count, thresh, need_eq);

  dim3 g3(64, B_IMG);
  select_mark<<<g3, 256, 0, stream>>>(cand, cand_count, thresh, eq_cnt, eqlist,
                                      sel, sel_count, out_mask);
  finalize_eq<<<B_IMG, EQCAP, 0, stream>>>(eqlist, eq_cnt, thresh, need_eq,
                                           sel, sel_count, out_mask);

  dim3 g5(64, B_IMG);
  gauss_scatter<<<g5, 256, 0, stream>>>(sel, sel_count, out_x);
  gauss_clip<<<g5, 256, 0, stream>>>(sel, sel_count, out_x);
}